// _Conv2dUpsampleDownsample_23845658427586
// MI455X (gfx1250) — compile-verified
//
#include <hip/hip_runtime.h>
#include <hip/hip_bf16.h>
#include <stdint.h>
#include <stddef.h>

typedef __attribute__((ext_vector_type(2))) float v2f;
typedef __attribute__((ext_vector_type(4))) float v4f;
typedef __attribute__((ext_vector_type(8))) float v8f;
typedef __attribute__((ext_vector_type(4))) unsigned int v4u;
typedef __attribute__((ext_vector_type(8))) int v8i;
typedef __attribute__((ext_vector_type(4))) int v4i;

#define NB   16      // batch
#define IC   256     // in channels
#define OC   256     // out channels
#define HW   64      // input spatial
#define XPD  66      // padded input spatial (64 + 1 left + 1 right)
#define YPD  131     // padded y spatial (129 + 1 left + 1 right)

// ---------------------------------------------------------------------------
// Tap tables: transposed-conv parity decomposition.
// Per-dim: even outputs (p=0) use taps (a=0,d=-1),(a=2,d=0); odd (p=1): (a=1,d=0)
// cls = pu*2+pv,  taps = cross product of row-taps x col-taps
// ---------------------------------------------------------------------------
__constant__ int c_ntap[4]    = {4, 2, 2, 1};
__constant__ int c_tapA[4][4] = {{0,0,2,2}, {0,2,0,0}, {1,1,0,0}, {1,0,0,0}};
__constant__ int c_tapB[4][4] = {{0,2,0,2}, {1,1,0,0}, {0,2,0,0}, {1,0,0,0}};
__constant__ int c_tapDi[4][4]= {{-1,-1,0,0},{-1,0,0,0},{0,0,0,0},{0,0,0,0}};
__constant__ int c_tapDj[4][4]= {{-1,0,-1,0},{0,0,0,0},{-1,0,0,0},{0,0,0,0}};

// ---------------------------------------------------------------------------
// Prep 1: swizzle weight (OIHW 256x256x3x3) into per-lane WMMA A-fragment order.
// Asw[tap9][mtile(16)][t(32)][lane(32)][d(4)] ;  o = 16*mt + lane%16,
// channel c = 8t + 4*(lane/16) + d   (K-slot permutation shared with B loads)
// ---------------------------------------------------------------------------
__global__ __launch_bounds__(256) void k_prep_w(const float* __restrict__ w,
                                                float* __restrict__ Asw) {
  int idx  = blockIdx.x * 256 + threadIdx.x;     // 9*16*32*32*4 = 589824 exact
  int d    = idx & 3;
  int lane = (idx >> 2) & 31;
  int t    = (idx >> 7) & 31;
  int mtg  = (idx >> 12) & 15;
  int tap  = idx >> 16;                          // 0..8
  int a = tap / 3, b = tap % 3;
  int o = mtg * 16 + (lane & 15);
  int c = 8 * t + 4 * (lane >> 4) + d;
  Asw[idx] = w[((o * IC + c) * 3 + a) * 3 + b];
}

// ---------------------------------------------------------------------------
// Prep 2: NCHW -> zero-padded NHWC  Xp[n][66][66][256]
// ---------------------------------------------------------------------------
__global__ __launch_bounds__(256) void k_prep_x(const float* __restrict__ x,
                                                float* __restrict__ Xp) {
  int idx = blockIdx.x * 256 + threadIdx.x;      // 16*66*66*256 = 17842176 exact
  int c  = idx & 255;
  int r  = idx >> 8;
  int j  = r % XPD;  r /= XPD;
  int i  = r % XPD;  r /= XPD;
  int n  = r;
  int ri = i - 1, rj = j - 1;
  float v = 0.0f;
  if (ri >= 0 && ri < HW && rj >= 0 && rj < HW)
    v = x[(((size_t)(n * IC + c) * HW) + ri) * HW + rj];
  Xp[idx] = v;
}

// ---------------------------------------------------------------------------
// Prep 3: zero the 1-element rim of Yb[n][o][131][131] (ws is poisoned)
// ---------------------------------------------------------------------------
__global__ __launch_bounds__(256) void k_zero_rims(float* __restrict__ Yb) {
  int idx = blockIdx.x * 256 + threadIdx.x;      // 16*256*520 = 2129920 exact
  int k  = idx % 520;
  int no = idx / 520;
  int up, vp;
  if (k < 262) {                 // rows 0 and 130, all 131 cols
    up = (k < 131) ? 0 : 130;
    vp = (k < 131) ? k : k - 131;
  } else {                       // cols 0 and 130, rows 1..129
    int k2 = k - 262;            // 0..257
    vp = (k2 < 129) ? 0 : 130;
    up = 1 + ((k2 < 129) ? k2 : k2 - 129);
  }
  Yb[((size_t)no * YPD + up) * YPD + vp] = 0.0f;
}

// ---------------------------------------------------------------------------
// TDM: 1-D contiguous tile, global -> LDS.  D# per CDNA5 ISA §8.3/§8.4:
// group0: count=1 | lds_addr | global_addr[56:0] | type=2
// group1: data_size=4B, tensor_dim0 = tile_dim0 = nelem (32-bit units)
// groups 2/3 (+ extra group of the 6-arg builtin): zero (<=2-D tensor)
// ---------------------------------------------------------------------------
__device__ inline void tdm_load_1d(unsigned int lds_off, const void* gsrc,
                                   unsigned int nelem) {
  unsigned long long ga = (unsigned long long)(uintptr_t)gsrc;
  v4u g0;
  g0[0] = 1u;                                            // count=1, no gather
  g0[1] = lds_off;                                       // lds_addr (bytes)
  g0[2] = (unsigned int)ga;                              // global_addr[31:0]
  g0[3] = (unsigned int)((ga >> 32) & 0x01FFFFFFu)       // global_addr[56:32]
        | (2u << 30);                                    // type=2 ("image")
  v8i g1 = {0, 0, 0, 0, 0, 0, 0, 0};
  g1[0] = (int)(2u << 16);                               // data_size=2 -> 4 bytes
  g1[1] = (int)((nelem & 0xFFFFu) << 16);                // tensor_dim0[15:0]  @ bits 63:48
  g1[2] = (int)((nelem >> 16) & 0xFFFFu);                // tensor_dim0[31:16] @ bits 79:64
  g1[3] = (int)((nelem & 0xFFFFu) << 16);                // tile_dim0 @ bits 127:112
  v4i z4 = {0, 0, 0, 0};
  v8i z8 = {0, 0, 0, 0, 0, 0, 0, 0};
  __builtin_amdgcn_tensor_load_to_lds(g0, g1, z4, z4, z8, 0);
}

// ---------------------------------------------------------------------------
// Stage 1: implicit-GEMM transposed conv via V_WMMA_F32_16X16X4_F32.
// block = 128 threads (4 waves). Wave wv: outC range [64*wv, 64*wv+64).
// Per tap: wave 0 TDM-stages the 32-pixel x 256-ch B row segment (32 KB,
// contiguous in NHWC Xp) into LDS; all 4 waves consume it via ds_load_b128.
// ---------------------------------------------------------------------------
__device__ inline v8f wmma4(v2f a, v2f b, v8f c) {
  return __builtin_amdgcn_wmma_f32_16x16x4_f32(false, a, false, b,
                                               (short)0, c, false, false);
}
__device__ inline v2f lo2(v4f v) { return __builtin_shufflevector(v, v, 0, 1); }
__device__ inline v2f hi2(v4f v) { return __builtin_shufflevector(v, v, 2, 3); }

__global__ __launch_bounds__(128) void k_conv_up(const float* __restrict__ Xp,
                                                 const float* __restrict__ Asw,
                                                 float* __restrict__ Yb) {
  __shared__ float lds_b[32 * IC];     // 32 pixels x 256 channels = 32 KB

  const int cg  = blockIdx.x;          // column group (32 class-cols)
  const int m   = blockIdx.y;          // class row
  const int z   = blockIdx.z;
  const int cls = z & 3;
  const int n   = z >> 2;
  const int pu  = cls >> 1, pv = cls & 1;
  const int rows = 65 - pu;
  const int cols = 65 - pv;
  if (m >= rows) return;               // uniform per block

  const int wv   = threadIdx.x >> 5;
  const int lane = threadIdx.x & 31;
  const int lm   = lane & 15;
  const int hl   = lane >> 4;
  const int l0   = cg * 32;

  v8f acc[4][2];
  #pragma unroll
  for (int a = 0; a < 4; ++a)
    #pragma unroll
    for (int b = 0; b < 2; ++b) acc[a][b] = {};

  // LDS base per lane: B layout [pixel(32)][channel(256)]
  const float* lb0 = lds_b + (size_t)lm * IC + 4 * hl;          // N-tile 0
  const float* lb1 = lds_b + (size_t)(lm + 16) * IC + 4 * hl;   // N-tile 1

  const int ntap = c_ntap[cls];
  for (int tp = 0; tp < ntap; ++tp) {
    const int ta   = c_tapA[cls][tp];
    const int tb   = c_tapB[cls][tp];
    const int di   = c_tapDi[cls][tp];
    const int dj   = c_tapDj[cls][tp];
    const int tap9 = ta * 3 + tb;

    // A fragment pointers: sequential b128 per t
    const v4f* ap[4];
    #pragma unroll
    for (int mt = 0; mt < 4; ++mt)
      ap[mt] = (const v4f*)(Asw +
               ((((size_t)tap9 * 16 + (wv * 4 + mt)) * 32) * 32 + lane) * 4);

    // --- stage B segment: one TDM per block per tap, shared by all 4 waves ---
    if (tp > 0) __syncthreads();       // previous tap's LDS reads complete
    if (wv == 0) {
      const int ii  = m + di + 1;      // padded row in [0,65]
      const int jj0 = l0 + dj + 1;     // padded first col
      const float* gsrc = Xp + ((size_t)(n * XPD + ii) * XPD + jj0) * IC;
      tdm_load_1d((unsigned int)(uintptr_t)lds_b, gsrc, 32 * IC);
      __builtin_amdgcn_s_wait_tensorcnt(0);
    }
    __syncthreads();                   // B tile visible to all waves

    #pragma unroll 2
    for (int t = 0; t < 32; ++t) {     // 2 K-steps (8 channels) per iter
      v4f A[4];
      #pragma unroll
      for (int mt = 0; mt < 4; ++mt) A[mt] = ap[mt][t * 32];
      v4f B0 = *(const v4f*)(lb0 + t * 8);   // ds_load_b128
      v4f B1 = *(const v4f*)(lb1 + t * 8);

      // K-step 2t : dwords {0,1}
      {
        v2f b0 = lo2(B0), b1 = lo2(B1);
        #pragma unroll
        for (int mt = 0; mt < 4; ++mt) {
          v2f av = lo2(A[mt]);
          acc[mt][0] = wmma4(av, b0, acc[mt][0]);
          acc[mt][1] = wmma4(av, b1, acc[mt][1]);
        }
      }
      // K-step 2t+1 : dwords {2,3}
      {
        v2f b0 = hi2(B0), b1 = hi2(B1);
        #pragma unroll
        for (int mt = 0; mt < 4; ++mt) {
          v2f av = hi2(A[mt]);
          acc[mt][0] = wmma4(av, b0, acc[mt][0]);
          acc[mt][1] = wmma4(av, b1, acc[mt][1]);
        }
      }
    }
  }

  // Store C tiles: lane holds col N=lm; v8f element g -> M = g + 8*hl
  const int u = 2 * m + pu;
  #pragma unroll
  for (int nt = 0; nt < 2; ++nt) {
    const int l = l0 + nt * 16 + lm;
    if (l >= cols) continue;
    const int v = 2 * l + pv;
    #pragma unroll
    for (int mt = 0; mt < 4; ++mt) {
      const int o = wv * 64 + mt * 16 + hl * 8;
      float* yp = Yb + ((size_t)(n * OC + o) * YPD + (u + 1)) * YPD + (v + 1);
      #pragma unroll
      for (int g = 0; g < 8; ++g)
        yp[(size_t)g * YPD * YPD] = acc[mt][nt][g];
    }
  }
}

// ---------------------------------------------------------------------------
// Stage 2: depthwise 4x4 FIR (separable [1,3,3,1]/4 per dim) + bias.
// One thread -> 4 consecutive outputs (float4 store). Pure bandwidth.
// ---------------------------------------------------------------------------
__global__ __launch_bounds__(256) void k_fir_bias(const float* __restrict__ Yb,
                                                  const float* __restrict__ bias,
                                                  float* __restrict__ out) {
  const size_t tid = (size_t)blockIdx.x * 256 + threadIdx.x;  // 16,777,216 exact
  const int t4 = (int)(tid & 31);
  const int s  = (int)((tid >> 5) & 127);
  const int o  = (int)((tid >> 12) & 255);
  const int n  = (int)(tid >> 20);
  const float bv = bias[o];
  const float fp[4] = {0.25f, 0.75f, 0.75f, 0.25f};
  const int t0 = t4 * 4;
  const float* yb = Yb + ((size_t)(n * OC + o) * YPD + s) * YPD + t0;
  float a0 = bv, a1 = bv, a2 = bv, a3 = bv;
  #pragma unroll
  for (int p = 0; p < 4; ++p) {
    const float* r = yb + (size_t)p * YPD;
    float rr[7];
    #pragma unroll
    for (int q = 0; q < 7; ++q) rr[q] = r[q];
    const float w = fp[p];
    #pragma unroll
    for (int q = 0; q < 4; ++q) {
      const float c = w * fp[q];
      a0 = fmaf(c, rr[q + 0], a0);
      a1 = fmaf(c, rr[q + 1], a1);
      a2 = fmaf(c, rr[q + 2], a2);
      a3 = fmaf(c, rr[q + 3], a3);
    }
  }
  float* op = out + ((size_t)(n * OC + o) * 128 + s) * 128 + t0;
  v4f res = {a0, a1, a2, a3};
  *(v4f*)op = res;
}

// ---------------------------------------------------------------------------
extern "C" void kernel_launch(void* const* d_in, const int* in_sizes, int n_in,
                              void* d_out, int out_size, void* d_ws, size_t ws_size,
                              hipStream_t stream) {
  (void)in_sizes; (void)n_in; (void)out_size; (void)ws_size;
  const float* x    = (const float*)d_in[0];
  const float* w    = (const float*)d_in[1];
  const float* bias = (const float*)d_in[2];
  float* out = (float*)d_out;

  char* ws = (char*)d_ws;
  float* Asw = (float*)ws;                              //  2,359,296 B
  float* Xp  = (float*)(ws + 2359296);                  // 71,368,704 B
  float* Yb  = (float*)(ws + 2359296 + 71368704);       // 281,165,824 B

  k_prep_w   <<<2304,  256, 0, stream>>>(w, Asw);
  k_prep_x   <<<69696, 256, 0, stream>>>(x, Xp);
  k_zero_rims<<<8320,  256, 0, stream>>>(Yb);
  dim3 g1(3, 65, 64);
  k_conv_up  <<<g1,    128, 0, stream>>>(Xp, Asw, Yb);
  k_fir_bias <<<65536, 256, 0, stream>>>(Yb, bias, out);
}